// JpegBasic_17231408792202
// MI455X (gfx1250) — compile-verified
//
#include <hip/hip_runtime.h>

// CDNA5 / gfx1250: wave32, WMMA f32 16x16x4
typedef float v2f __attribute__((ext_vector_type(2)));
typedef float v8f __attribute__((ext_vector_type(8)));

__device__ __constant__ float LUMA_TBL[64] = {
  16, 11, 10, 16, 24, 40, 51, 61,
  12, 12, 14, 19, 26, 58, 60, 55,
  14, 13, 16, 24, 40, 57, 69, 56,
  14, 17, 22, 29, 51, 87, 80, 62,
  18, 22, 37, 56, 68,109,103, 77,
  24, 35, 55, 64, 81,104,113, 92,
  49, 64, 78, 87,103,121,120,101,
  72, 92, 95, 98,112,100,103, 99
};
__device__ __constant__ float CHROMA_TBL[64] = {
  17, 18, 24, 47, 99, 99, 99, 99,
  18, 21, 26, 66, 99, 99, 99, 99,
  24, 26, 56, 99, 99, 99, 99, 99,
  47, 66, 99, 99, 99, 99, 99, 99,
  99, 99, 99, 99, 99, 99, 99, 99,
  99, 99, 99, 99, 99, 99, 99, 99,
  99, 99, 99, 99, 99, 99, 99, 99,
  99, 99, 99, 99, 99, 99, 99, 99
};

#define LSTR 18   // padded LDS row stride (floats): conflict-free for both read patterns

__device__ __forceinline__ float dmat(int i, int j) {
  // 8x8 DCT-II matrix entry
  if (i == 0) return 0.35355339059327373f;                       // sqrt(1/8)
  return 0.5f * __cosf(0.19634954084936207f * (float)(i * (2 * j + 1))); // cos(pi*i*(2j+1)/16)*sqrt(2/8)
}

// 16x16x16 f32 matmul as 4 chained v_wmma_f32_16x16x4_f32
__device__ __forceinline__ v8f wmma16(const float a[8], const float b[8]) {
  v8f acc = {};
#pragma unroll
  for (int kk = 0; kk < 4; ++kk) {
    v2f av = { a[2 * kk], a[2 * kk + 1] };
    v2f bv = { b[2 * kk], b[2 * kk + 1] };
    acc = __builtin_amdgcn_wmma_f32_16x16x4_f32(false, av, false, bv,
                                                (short)0, acc, false, false);
  }
  return acc;
}

__device__ __forceinline__ void lds_wait() {
  __builtin_amdgcn_wave_barrier();
  asm volatile("s_wait_dscnt 0" ::: "memory");   // CDNA5 split DS counter wait
  __builtin_amdgcn_wave_barrier();
}

// store C/D-layout 16x16 tile: lane(n,h) slot v holds M[v+8h][n]
__device__ __forceinline__ void lds_store_c(float* L, v8f acc, int n, int h) {
#pragma unroll
  for (int v = 0; v < 8; ++v) L[(v + 8 * h) * LSTR + n] = acc[v];
}
// load B-layout of the SAME matrix: reg q -> M[4*(q>>1)+2h+(q&1)][n]
__device__ __forceinline__ void lds_load_b(const float* L, float* bB, int n, int h) {
#pragma unroll
  for (int q = 0; q < 8; ++q) {
    int r = 4 * (q >> 1) + 2 * h + (q & 1);
    bB[q] = L[r * LSTR + n];
  }
}
// load B-layout of the TRANSPOSE: reg q -> M[n][4*(q>>1)+2h+(q&1)]
__device__ __forceinline__ void lds_load_bt(const float* L, float* bB, int n, int h) {
#pragma unroll
  for (int q = 0; q < 8; ++q) {
    int r = 4 * (q >> 1) + 2 * h + (q & 1);
    bB[q] = L[n * LSTR + r];
  }
}

// One channel through DCT -> quant/dequant -> IDCT. xB in B-layout, result in C-layout.
__device__ __forceinline__ void process_channel(const float xB[8], const float aD[8],
                                                const float aDt[8], const float qt[8],
                                                float* L, int n, int h, float outv[8]) {
  // pass 1: S = Dbig @ X                     (C-layout)
  v8f acc = wmma16(aD, xB);
  lds_store_c(L, acc, n, h);
  lds_wait();
  float tB[8];
  lds_load_bt(L, tB, n, h);                   // S^T in B-layout
  // pass 2: F^T = Dbig @ S^T                 (C-layout; F = Dbig X Dbig^T)
  acc = wmma16(aD, tB);
  // quantize / dequantize (F^T[m][n] = dct[u=n][v=m], table[u&7][v])
#pragma unroll
  for (int v = 0; v < 8; ++v) {
    float tq = qt[v];
    acc[v] = rintf(acc[v] / tq) * tq;         // RNE, matches jnp.round
  }
  lds_store_c(L, acc, n, h);
  lds_wait();
  lds_load_b(L, tB, n, h);                    // F^T in B-layout
  // pass 3: W = Dbig^T @ F^T                 (C-layout)
  acc = wmma16(aDt, tB);
  lds_store_c(L, acc, n, h);
  lds_wait();
  lds_load_bt(L, tB, n, h);                   // W^T in B-layout
  // pass 4: G = Dbig^T @ W^T = Dbig^T F Dbig (C-layout = spatial pixels)
  acc = wmma16(aDt, tB);
#pragma unroll
  for (int v = 0; v < 8; ++v) outv[v] = acc[v];
}

__global__ __launch_bounds__(256) void jpeg_wmma_kernel(
    const float* __restrict__ img, const int* __restrict__ subsample_p,
    const int* __restrict__ scale_p, float* __restrict__ out, int nwaves) {
  __shared__ float lds[8 * 16 * LSTR];
  const int lane = threadIdx.x & 31;
  const int wid  = threadIdx.x >> 5;
  float* L = &lds[wid * 16 * LSTR];

  const int w = blockIdx.x * 8 + wid;         // one wave == one 16x16 pixel tile
  if (w >= nwaves) return;
  const int b  = w >> 10;                     // 1024 tiles per 512x512 image
  const int t  = w & 1023;
  const int ty = t >> 5, tx = t & 31;

  const int n = lane & 15;                    // column owned by this lane (B and C layouts)
  const int h = lane >> 4;                    // wave half

  const int   subsample = subsample_p[0];
  const float sf        = (float)scale_p[0];

  // Constant A-operands: Dbig = diag(D,D) and its transpose, in WMMA A-layout chunks.
  // A-layout: lane row m = lane&15; chunk kk reg j -> column k = 4kk+2h+j.
  float aD[8], aDt[8];
  const int m = n;
#pragma unroll
  for (int q = 0; q < 8; ++q) {
    int k = 4 * (q >> 1) + 2 * h + (q & 1);
    bool diag = (m >> 3) == (k >> 3);
    aD[q]  = diag ? dmat(m & 7, k & 7) : 0.0f;
    aDt[q] = diag ? dmat(k & 7, m & 7) : 0.0f;
  }

  // Per-lane quant tables: slot v quantizes F^T[v+8h][n] = dct[u=n][v] -> tbl[n&7][v]
  float qY[8], qC[8];
#pragma unroll
  for (int v = 0; v < 8; ++v) {
    qY[v] = fmaxf(1.0f, rintf(LUMA_TBL[(n & 7) * 8 + v] * sf));
    qC[v] = fmaxf(1.0f, rintf(CHROMA_TBL[(n & 7) * 8 + v] * sf));
  }

  // Load RGB once, directly in B-layout, convert to YUV.
  const long plane = 512L * 512L;
  const float* base = img + (long)b * 3 * plane;
  const int col = tx * 16 + n;
  float yB[8], uB[8], vB[8];
#pragma unroll
  for (int q = 0; q < 8; ++q) {
    int r = 4 * (q >> 1) + 2 * h + (q & 1);
    long idx = (long)(ty * 16 + r) * 512 + col;
    float rr = base[idx];
    float gg = base[plane + idx];
    float bb = base[2 * plane + idx];
    rr = (fminf(fmaxf(rr, -1.0f), 1.0f) + 1.0f) * 127.5f;
    gg = (fminf(fmaxf(gg, -1.0f), 1.0f) + 1.0f) * 127.5f;
    bb = (fminf(fmaxf(bb, -1.0f), 1.0f) + 1.0f) * 127.5f;
    yB[q] =  0.299f   * rr + 0.587f  * gg + 0.114f * bb;
    uB[q] = -0.1687f  * rr - 0.3313f * gg + 0.5f   * bb;
    vB[q] =  0.5f     * rr - 0.4187f * gg - 0.0813f* bb;
  }

  // Chroma 2x subsample: value(r,n) <- value(r&~1, n&~1).
  // In B-layout: row part = local reg j=0 of same chunk; col part = lane&~1 shuffle.
  if (subsample == 2) {
#pragma unroll
    for (int kk = 0; kk < 4; ++kk) {
      float us = __shfl(uB[2 * kk], lane & ~1, 32);
      float vs = __shfl(vB[2 * kk], lane & ~1, 32);
      uB[2 * kk] = us; uB[2 * kk + 1] = us;
      vB[2 * kk] = vs; vB[2 * kk + 1] = vs;
    }
  }

  float oy[8], ou[8], ov[8];
  process_channel(yB, aD, aDt, qY, L, n, h, oy);
  process_channel(uB, aD, aDt, qC, L, n, h, ou);
  process_channel(vB, aD, aDt, qC, L, n, h, ov);

  // YUV -> RGB, rescale, store (C-layout: slot v -> pixel (row v+8h, col n), coalesced)
  float* ob = out + (long)b * 3 * plane;
#pragma unroll
  for (int v = 0; v < 8; ++v) {
    float yy = oy[v], uu = ou[v], vv = ov[v];
    float rr = yy + 1.40198758f * vv;
    float gg = yy - 0.344113281f * uu - 0.714103821f * vv;
    float bb = yy + 1.77197812f * uu;
    long idx = (long)(ty * 16 + v + 8 * h) * 512 + col;
    const float s = 2.0f / 255.0f;
    ob[idx]             = rr * s - 1.0f;
    ob[plane + idx]     = gg * s - 1.0f;
    ob[2 * plane + idx] = bb * s - 1.0f;
  }
}

extern "C" void kernel_launch(void* const* d_in, const int* in_sizes, int n_in,
                              void* d_out, int out_size, void* d_ws, size_t ws_size,
                              hipStream_t stream) {
  const float* img = (const float*)d_in[0];
  const int* sub   = (const int*)d_in[1];
  const int* sc    = (const int*)d_in[2];
  float* out       = (float*)d_out;

  const int B = in_sizes[0] / (3 * 512 * 512);   // 32
  const int nwaves = B * 1024;                   // one wave per 16x16 tile
  const int nblocks = (nwaves + 7) / 8;          // 8 waves (256 thr) per block
  hipLaunchKernelGGL(jpeg_wmma_kernel, dim3(nblocks), dim3(256), 0, stream,
                     img, sub, sc, out, nwaves);
}